// attn_764504178974
// MI455X (gfx1250) — compile-verified
//
#include <hip/hip_runtime.h>
#include <math.h>

#define B_    1024
#define L_    512
#define ATTN_ 1000
#define HID_  1000
#define K_    (HID_ + ATTN_)   // 2000
#define XSTRIDE 2048
#define TL    4                // rows of ho per LDS tile (double buffered)
#define NCHUNK (L_ / TL)       // 128

typedef __attribute__((ext_vector_type(2))) float v2f;
typedef __attribute__((ext_vector_type(8))) float v8f;

// exact pointee type expected by the async builtin: int __vector_size__(16)
typedef int v4i __attribute__((__vector_size__(4 * sizeof(int))));
typedef __attribute__((address_space(1))) v4i gv4i;   // global int4
typedef __attribute__((address_space(3))) v4i sv4i;   // LDS int4

#if defined(__has_builtin)
#if __has_builtin(__builtin_amdgcn_global_load_async_to_lds_b128) && \
    __has_builtin(__builtin_amdgcn_s_wait_asynccnt)
#define HAVE_ASYNC 1
#endif
#endif

#ifdef HAVE_ASYNC
#define ASYNC_WAIT(n) __builtin_amdgcn_s_wait_asynccnt(n)
#else
#define ASYNC_WAIT(n)
#endif

// ---------------------------------------------------------------------------
// Kernel 1: fused scores -> online softmax -> context, one block per batch b.
// Reads ho exactly once, streamed into LDS through the CDNA5 async
// global->LDS path (no VGPR round trip), double buffered so chunk c+1's HBM
// traffic overlaps chunk c's compute. Also stages hc into X.
// X layout: [1024][2048] f32 in d_ws; cols 0..999 = hc, 1000..1999 = context.
// ---------------------------------------------------------------------------
__global__ __launch_bounds__(256) void attn_ctx_kernel(
    const float* __restrict__ hc, const float* __restrict__ ho,
    const unsigned char* __restrict__ mask, const float* __restrict__ Wa_w,
    const float* __restrict__ Wa_b, float* __restrict__ X)
{
    __shared__ __align__(16) float tile[2][TL * ATTN_];  // 2 x 16000 B
    __shared__ float wa_lds[ATTN_];                      // 4000 B
    __shared__ float wavePart[8];

    const int b    = blockIdx.x;
    const int tid  = threadIdx.x;
    const int wave = tid >> 5;
    const int lane = tid & 31;

    // stage hc[b] -> X[b][0:1000] (250 float4 transfers, 16B aligned)
    if (tid < 250) {
        const float4 v = *(const float4*)(hc + (size_t)b * HID_ + 4 * tid);
        *(float4*)(X + (size_t)b * XSTRIDE + 4 * tid) = v;
    }
    // stage Wa_w into LDS (visible after first S1 barrier)
    for (int d = tid; d < ATTN_; d += 256) wa_lds[d] = Wa_w[d];
    const float wab = Wa_b[0];

    // each thread owns 4 context columns: tid, tid+256, tid+512, tid+768
    const int d1 = tid + 256, d2 = tid + 512, d3 = tid + 768;
    float acc0 = 0.f, acc1 = 0.f, acc2 = 0.f, acc3 = 0.f;
    float m = -INFINITY, s = 0.f;

    const float*         horow = ho   + (size_t)b * L_ * ATTN_;
    const unsigned char* mrow  = mask + (size_t)b * L_;

    // issue one chunk (TL rows x 1000 f32) into tile[buf] via async b128 copies
    auto issue_chunk = [&](int buf, int l0) {
        if (tid < 250) {
            #pragma unroll
            for (int r = 0; r < TL; ++r) {
                const float* src = horow + (size_t)(l0 + r) * ATTN_ + 4 * tid;
                float*       dst = &tile[buf][r * ATTN_ + 4 * tid];
#ifdef HAVE_ASYNC
                __builtin_amdgcn_global_load_async_to_lds_b128(
                    (gv4i*)const_cast<float*>(src),
                    (sv4i*)dst, /*imm offset*/0, /*cpol*/0);
#else
                *(float4*)dst = *(const float4*)src;
#endif
            }
        }
    };

    issue_chunk(0, 0);   // prologue: chunk 0 in flight

    for (int c = 0; c < NCHUNK; ++c) {
        const int nb = c & 1;
        if (c + 1 < NCHUNK) {
            issue_chunk(1 - nb, (c + 1) * TL);  // overlap next chunk's HBM traffic
            ASYNC_WAIT(TL);   // async loads retire in order: <=TL left => chunk c landed
        } else {
            ASYNC_WAIT(0);
        }
        __syncthreads();      // S1: tile[nb] fully written by all waves

        // phase A: 8 waves cover TL=4 rows x 2 halves of d
        {
            const int r     = wave & 3;
            const int dbase = (wave >> 2) * 500;
            const float* trow = &tile[nb][r * ATTN_];
            float p = 0.f;
            for (int i = lane; i < 500; i += 32)
                p += trow[dbase + i] * wa_lds[dbase + i];
            #pragma unroll
            for (int off = 16; off; off >>= 1) p += __shfl_xor(p, off, 32);
            if (lane == 0) wavePart[wave] = p;
        }
        __syncthreads();      // S2: wavePart ready

        // phase B: online softmax update (identical on all threads)
        {
            const int l0 = c * TL;
            const uchar4 mv = *(const uchar4*)(mrow + l0);  // l0 % 4 == 0
            const unsigned char mk[TL] = { mv.x, mv.y, mv.z, mv.w };
            float sc[TL], pk[TL];
            float cmax = -INFINITY;
            #pragma unroll
            for (int r = 0; r < TL; ++r) {
                float v = wavePart[r] + wavePart[r + 4] + wab;
                if (mk[r]) v = -10000.0f;
                sc[r] = v;
                cmax  = fmaxf(cmax, v);
            }
            const float new_m = fmaxf(m, cmax);
            const float scale = __expf(m - new_m);  // m == -inf first chunk -> 0
            float psum = 0.f;
            #pragma unroll
            for (int r = 0; r < TL; ++r) { pk[r] = __expf(sc[r] - new_m); psum += pk[r]; }
            s = s * scale + psum;
            m = new_m;
            acc0 *= scale; acc1 *= scale; acc2 *= scale; acc3 *= scale;
            #pragma unroll
            for (int r = 0; r < TL; ++r) {
                const float* trow = &tile[nb][r * ATTN_];
                acc0 += pk[r] * trow[tid];
                if (d1 < ATTN_) acc1 += pk[r] * trow[d1];
                if (d2 < ATTN_) acc2 += pk[r] * trow[d2];
                if (d3 < ATTN_) acc3 += pk[r] * trow[d3];
            }
        }
        __syncthreads();      // S3: tile[nb]/wavePart free for reuse
    }

    const float inv = 1.0f / s;
    float* xc = X + (size_t)b * XSTRIDE + HID_;
    xc[tid] = acc0 * inv;
    if (d1 < ATTN_) xc[d1] = acc1 * inv;
    if (d2 < ATTN_) xc[d2] = acc2 * inv;
    if (d3 < ATTN_) xc[d3] = acc3 * inv;
}

// ---------------------------------------------------------------------------
// Kernel 2: out[b][o] = sum_k X[b][k] * Wc_w[o][k] + Wc_b[o]
// WMMA f32 16x16x4, one 16x16 output tile per wave, K = 2000 in steps of 4.
// A frag (16x4): lanes 0-15 -> K0/K1, lanes 16-31 -> K2/K3  (float2 loads)
// B frag (4x16): same half-wave K split, lane%16 = column     (float2 loads)
// D: lane%16 = column, VGPR r = row r (+8 for upper half-wave)
// ---------------------------------------------------------------------------
__global__ __launch_bounds__(256) void out_gemm_kernel(
    const float* __restrict__ X,     // [1024][2048]
    const float* __restrict__ W,     // [1000][2000]
    const float* __restrict__ bias,  // [1000]
    float* __restrict__ out)         // [1024][1000]
{
    const int tileId = blockIdx.x * 8 + (threadIdx.x >> 5);  // 0..4031
    const int tm   = tileId & 63;    // 64 tiles along M (1024/16)
    const int tn   = tileId >> 6;    // 63 tiles along N (ceil(1000/16))
    const int lane = threadIdx.x & 31;
    const int half = lane >> 4;      // K half: 0 -> K0/K1, 1 -> K2/K3
    const int r16  = lane & 15;

    const int m  = tm * 16 + r16;
    const int o  = tn * 16 + r16;
    const int oc = (o < HID_) ? o : (HID_ - 1);              // clamp edge loads

    const float* arow = X + (size_t)m  * XSTRIDE + half * 2; // 8B aligned
    const float* brow = W + (size_t)oc * K_      + half * 2; // 8000B row stride, 8B aligned

    v8f acc = {};
    for (int k = 0; k < K_; k += 4) {
        const v2f a  = *(const v2f*)(arow + k);
        const v2f bb = *(const v2f*)(brow + k);
        acc = __builtin_amdgcn_wmma_f32_16x16x4_f32(
            /*neg_a=*/false, a, /*neg_b=*/false, bb,
            /*c_mod=*/(short)0, acc, /*reuse_a=*/false, /*reuse_b=*/false);
    }

    if (o < HID_) {
        const int   mbase = tm * 16 + half * 8;
        const float bv    = bias[o];
        #pragma unroll
        for (int r = 0; r < 8; ++r)
            out[(size_t)(mbase + r) * HID_ + o] = acc[r] + bv;
    }
}

// ---------------------------------------------------------------------------
extern "C" void kernel_launch(void* const* d_in, const int* in_sizes, int n_in,
                              void* d_out, int out_size, void* d_ws, size_t ws_size,
                              hipStream_t stream) {
    const float*         hc   = (const float*)d_in[0];          // [1024,1000]
    const float*         ho   = (const float*)d_in[1];          // [1024,512,1000]
    const unsigned char* mask = (const unsigned char*)d_in[2];  // [1024,512] bool
    const float*         Wa_w = (const float*)d_in[3];          // [1,1000]
    const float*         Wa_b = (const float*)d_in[4];          // [1]
    const float*         Wc_w = (const float*)d_in[5];          // [1000,2000]
    const float*         Wc_b = (const float*)d_in[6];          // [1000]
    float*               out  = (float*)d_out;                  // [1024,1000]
    float*               X    = (float*)d_ws;                   // [1024,2048] = 8 MB

    attn_ctx_kernel<<<B_, 256, 0, stream>>>(hc, ho, mask, Wa_w, Wa_b, X);
    out_gemm_kernel<<<504, 256, 0, stream>>>(X, Wc_w, Wc_b, out);
}